// LDAHead_87763361726703
// MI455X (gfx1250) — compile-verified
//
#include <hip/hip_runtime.h>
#include <math.h>

namespace {

constexpr int kC      = 1000;
constexpr int kCPad   = 1008;   // 63 * 16 (class dim padded to WMMA tile)
constexpr int kD      = 64;
constexpr int kB      = 2048;
constexpr float kEpsStats = 1e-5f;
constexpr float kEpsPrec  = 1e-6f;

// workspace layout (float offsets)
constexpr int OFF_COUNTS = 0;                       // kC      eps-added class counts
constexpr int OFF_MEAN   = OFF_COUNTS + kC;         // kC*kD   class means
constexpr int OFF_VARSUM = OFF_MEAN   + kC * kD;    // kC*kD   per-class sum of squared residuals
constexpr int OFF_MUPAD  = OFF_VARSUM + kC * kD;    // kCPad*kD  precision-scaled means (B operand)
constexpr int OFF_PREC   = OFF_MUPAD  + kCPad * kD; // kD      diagonal precision
constexpr int OFF_TOTAL  = OFF_PREC   + kD;         // 1       sum of eps-added counts
constexpr int OFF_CCONST = OFF_TOTAL  + 1;          // kCPad   log(prior) - 0.5 * mu' P mu
constexpr int OFF_QH     = OFF_CCONST + kCPad;      // kB      -0.5 * z' P z
// total ~196.6K floats (~787 KB) of d_ws

typedef __attribute__((ext_vector_type(2))) float v2f;
typedef __attribute__((ext_vector_type(8))) float v8f;

// --- Pass 1: per-class counts + means (deterministic scan, one block per class) ---
__global__ __launch_bounds__(64)
void class_stats_kernel(const float* __restrict__ z, const int* __restrict__ y,
                        float* __restrict__ ws) {
  const int c = blockIdx.x;
  const int d = threadIdx.x;
  float s = 0.f, cnt = 0.f;
  for (int b = 0; b < kB; ++b) {
    if (y[b] == c) { s += z[b * kD + d]; cnt += 1.f; }
  }
  const float cnte = cnt + kEpsStats;
  if (d == 0) ws[OFF_COUNTS + c] = cnte;
  ws[OFF_MEAN + c * kD + d] = s / cnte;
}

// --- Pass 2: per-class centered sum-of-squares ---
__global__ __launch_bounds__(64)
void var_stats_kernel(const float* __restrict__ z, const int* __restrict__ y,
                      float* __restrict__ ws) {
  const int c = blockIdx.x;
  const int d = threadIdx.x;
  const float m = ws[OFF_MEAN + c * kD + d];
  float s = 0.f;
  for (int b = 0; b < kB; ++b) {
    if (y[b] == c) { const float t = z[b * kD + d] - m; s += t * t; }
  }
  ws[OFF_VARSUM + c * kD + d] = s;
}

// --- Pass 3: pooled covariance -> diagonal precision  (prior_c*var_c = varsum_c/total) ---
__global__ __launch_bounds__(64)
void precision_kernel(float* __restrict__ ws) {
  const int d = threadIdx.x;
  float total = 0.f;
  for (int c = 0; c < kC; ++c) total += ws[OFF_COUNTS + c];
  float s = 0.f;
  for (int c = 0; c < kC; ++c) s += ws[OFF_VARSUM + c * kD + d];
  const float pooled = s / total + kEpsStats;
  ws[OFF_PREC + d] = 1.0f / fmaxf(pooled, kEpsPrec);
  if (d == 0) ws[OFF_TOTAL] = total;
}

// --- Pass 4: B operand (prec*mu, zero-padded) + per-class constant ---
__global__ __launch_bounds__(64)
void mu_prep_kernel(float* __restrict__ ws) {
  const int c = blockIdx.x;
  const int d = threadIdx.x;
  __shared__ float red[kD];
  if (c >= kC) {                          // zero padding rows so WMMA reads are harmless
    ws[OFF_MUPAD + c * kD + d] = 0.f;
    if (d == 0) ws[OFF_CCONST + c] = 0.f;
    return;
  }
  const float p = ws[OFF_PREC + d];
  const float m = ws[OFF_MEAN + c * kD + d];
  const float v = p * m;
  ws[OFF_MUPAD + c * kD + d] = v;
  red[d] = v * m;
  __syncthreads();
  for (int stride = kD / 2; stride > 0; stride >>= 1) {
    if (d < stride) red[d] += red[d + stride];
    __syncthreads();
  }
  if (d == 0) {
    const float prior = ws[OFF_COUNTS + c] / ws[OFF_TOTAL];
    ws[OFF_CCONST + c] = logf(prior) - 0.5f * red[0];
  }
}

// --- Pass 5: per-row quadratic term qh[b] = -0.5 * z' P z ---
__global__ __launch_bounds__(256)
void row_quad_kernel(const float* __restrict__ z, float* __restrict__ ws) {
  __shared__ float pshared[kD];
  if (threadIdx.x < kD) pshared[threadIdx.x] = ws[OFF_PREC + threadIdx.x];
  __syncthreads();
  const int b = blockIdx.x * blockDim.x + threadIdx.x;
  if (b >= kB) return;
  float s = 0.f;
#pragma unroll
  for (int d = 0; d < kD; ++d) {
    const float t = z[b * kD + d];
    s += pshared[d] * t * t;
  }
  ws[OFF_QH + b] = -0.5f * s;
}

// --- Pass 6: scores via fp32 WMMA GEMM: out = z @ (P*mu)^T + cconst[c] + qh[b] ---
// One wave per 16x16 tile; 8 waves (256 threads) per block.
__global__ __launch_bounds__(256)
void lda_score_wmma_kernel(const float* __restrict__ z, const float* __restrict__ ws,
                           float* __restrict__ out) {
  const int wave = threadIdx.x >> 5;
  const int lane = threadIdx.x & 31;
  const int tile = blockIdx.x * 8 + wave;   // 128*63 = 8064 tiles total
  const int tm = tile & 127;                // M tile (kB/16 = 128)
  const int tn = tile >> 7;                 // N tile (0..62)
  const int row0 = tm * 16;
  const int col0 = tn * 16;
  const int half = lane >> 4;               // K sub-group select per ISA fragment layout
  const int lm   = lane & 15;

  // A fragment: A[M=lm][K = k + v + 2*half]  -> contiguous float2 per lane
  const float* __restrict__ ap = z + (row0 + lm) * kD + 2 * half;
  // B fragment: B[K][N=lm] = mupad[col0+lm][K] -> identical addressing on row-major mupad
  const float* __restrict__ bp = ws + OFF_MUPAD + (col0 + lm) * kD + 2 * half;

  v8f acc = {};
#pragma unroll
  for (int k = 0; k < kD; k += 4) {
    const v2f a = *(const v2f*)(ap + k);
    const v2f b = *(const v2f*)(bp + k);
    // D = A(16x4,f32) x B(4x16,f32) + C(16x16,f32)
    acc = __builtin_amdgcn_wmma_f32_16x16x4_f32(
        /*neg_a=*/false, a, /*neg_b=*/false, b,
        /*c_mod=*/(short)0, acc, /*reuse_a=*/false, /*reuse_b=*/false);
  }

  const int col = col0 + lm;
  const float cc = ws[OFF_CCONST + col];    // padded, safe read
  if (col < kC) {
#pragma unroll
    for (int r = 0; r < 8; ++r) {           // D layout: VGPR r -> M = r + 8*half, N = lane%16
      const int row = row0 + r + 8 * half;
      out[row * kC + col] = acc[r] + cc + ws[OFF_QH + row];
    }
  }
}

} // anonymous namespace

extern "C" void kernel_launch(void* const* d_in, const int* in_sizes, int n_in,
                              void* d_out, int out_size, void* d_ws, size_t ws_size,
                              hipStream_t stream) {
  (void)in_sizes; (void)n_in; (void)out_size; (void)ws_size;
  const float* z = (const float*)d_in[0];
  const int*   y = (const int*)d_in[1];
  float* ws  = (float*)d_ws;
  float* out = (float*)d_out;

  class_stats_kernel<<<kC, kD, 0, stream>>>(z, y, ws);
  var_stats_kernel<<<kC, kD, 0, stream>>>(z, y, ws);
  precision_kernel<<<1, kD, 0, stream>>>(ws);
  mu_prep_kernel<<<kCPad, kD, 0, stream>>>(ws);
  row_quad_kernel<<<kB / 256, 256, 0, stream>>>(z, ws);

  const int tiles = (kB / 16) * (kCPad / 16);   // 8064
  lda_score_wmma_kernel<<<tiles / 8, 256, 0, stream>>>(z, ws, out);
}